// SoftMaxCustom_46248207843553
// MI455X (gfx1250) — compile-verified
//
#include <hip/hip_runtime.h>

#define EPS_F 1e-5f

// ---- CDNA5 native f32 VMEM atomics (device scope), no CAS loops ----
__device__ __forceinline__ void gmin_f32(float* addr, float v) {
  asm volatile("global_atomic_min_num_f32 %0, %1, off scope:SCOPE_DEV"
               :: "v"(addr), "v"(v) : "memory");
}
__device__ __forceinline__ void gadd_f32(float* addr, float v) {
  asm volatile("global_atomic_add_f32 %0, %1, off scope:SCOPE_DEV"
               :: "v"(addr), "v"(v) : "memory");
}

// ws layout: [ seg_min : nseg*16 floats ][ seg_sum : nseg*16 floats ]
__global__ void seg_init(float* __restrict__ ws, const int* __restrict__ nseg_p) {
  const long long total = (long long)(*nseg_p) * 16;
  float* seg_min = ws;
  float* seg_sum = ws + total;
  for (long long i = blockIdx.x * (long long)blockDim.x + threadIdx.x;
       i < total; i += (long long)gridDim.x * blockDim.x) {
    seg_min[i] = __int_as_float(0x7f800000);  // +inf
    seg_sum[i] = 0.0f;
  }
}

// One thread per (edge, quad-of-4-features): b128 load + 4 f32-min atomics.
__global__ void seg_min_pass(const float4* __restrict__ data4,
                             const int* __restrict__ index,
                             float* __restrict__ seg_min,
                             long long nquads) {
  long long t = blockIdx.x * (long long)blockDim.x + threadIdx.x;
  if (t >= nquads) return;
  const long long e = t >> 2;
  const int q = (int)(t & 3);
  const int idx = index[e];
  const float4 v = data4[t];
  float* b = seg_min + ((long long)idx * 16 + q * 4);
  gmin_f32(b + 0, v.x);
  gmin_f32(b + 1, v.y);
  gmin_f32(b + 2, v.z);
  gmin_f32(b + 3, v.w);
}

// num = exp(data - seg_min[idx]) + eps ; store num ; atomic-add into seg_sum.
__global__ void seg_expsum_pass(const float4* __restrict__ data4,
                                const int* __restrict__ index,
                                const int* __restrict__ nseg_p,
                                float* __restrict__ ws,
                                float4* __restrict__ out4,
                                long long nquads) {
  long long t = blockIdx.x * (long long)blockDim.x + threadIdx.x;
  if (t >= nquads) return;
  const long long e = t >> 2;
  const int q = (int)(t & 3);
  const int idx = index[e];

  const float4* seg_min4 = (const float4*)ws;
  float* seg_sum = ws + (long long)(*nseg_p) * 16;

  const float4 v = data4[t];
  const float4 m = seg_min4[(long long)idx * 4 + q];

  float4 num;
  num.x = __expf(v.x - m.x) + EPS_F;
  num.y = __expf(v.y - m.y) + EPS_F;
  num.z = __expf(v.z - m.z) + EPS_F;
  num.w = __expf(v.w - m.w) + EPS_F;

  out4[t] = num;

  float* b = seg_sum + ((long long)idx * 16 + q * 4);
  gadd_f32(b + 0, num.x);
  gadd_f32(b + 1, num.y);
  gadd_f32(b + 2, num.z);
  gadd_f32(b + 3, num.w);
}

// out = num / seg_sum[idx]
__global__ void seg_div_pass(const int* __restrict__ index,
                             const int* __restrict__ nseg_p,
                             const float* __restrict__ ws,
                             float4* __restrict__ out4,
                             long long nquads) {
  long long t = blockIdx.x * (long long)blockDim.x + threadIdx.x;
  if (t >= nquads) return;
  const long long e = t >> 2;
  const int q = (int)(t & 3);
  const int idx = index[e];

  const float4* seg_sum4 = (const float4*)(ws + (long long)(*nseg_p) * 16);
  float4 num = out4[t];
  const float4 s = seg_sum4[(long long)idx * 4 + q];

  num.x = __fdividef(num.x, s.x);
  num.y = __fdividef(num.y, s.y);
  num.z = __fdividef(num.z, s.z);
  num.w = __fdividef(num.w, s.w);

  out4[t] = num;
}

extern "C" void kernel_launch(void* const* d_in, const int* in_sizes, int n_in,
                              void* d_out, int out_size, void* d_ws, size_t ws_size,
                              hipStream_t stream) {
  const float* data  = (const float*)d_in[0];   // [n_edges, 16] f32
  const int* index   = (const int*)d_in[1];     // [n_edges]
  const int* nseg_p  = (const int*)d_in[2];     // scalar on device
  float* ws          = (float*)d_ws;
  float4* out4       = (float4*)d_out;

  const long long n_edges = (long long)in_sizes[1];
  const long long nquads  = n_edges * 4;        // 4 float4 per edge row

  const int threads = 256;
  const unsigned blocks = (unsigned)((nquads + threads - 1) / threads);

  // 1) init segment tables (grid-stride; nseg read on device)
  seg_init<<<2048, threads, 0, stream>>>(ws, nseg_p);
  // 2) per-feature segment min
  seg_min_pass<<<blocks, threads, 0, stream>>>((const float4*)data, index, ws, nquads);
  // 3) exp(data - min) + eps -> out, and segment sums
  seg_expsum_pass<<<blocks, threads, 0, stream>>>((const float4*)data, index, nseg_p,
                                                  ws, out4, nquads);
  // 4) normalize
  seg_div_pass<<<blocks, threads, 0, stream>>>(index, nseg_p, ws, out4, nquads);
}